// GraphClassifier_74646531604699
// MI455X (gfx1250) — compile-verified
//
#include <hip/hip_runtime.h>
#include <math.h>

// ---------------------------------------------------------------------------
// GraphClassifier for MI455X (gfx1250, wave32).
// Dense node GEMMs -> V_WMMA_F32_16X16X4_F32 (fp32 matrix pipe).
// Edge phase (HBM-bound) -> wave-per-edge coalesced rows + shfl reductions
// + float atomics for segment softmax / segment sum.
// ---------------------------------------------------------------------------

typedef __attribute__((ext_vector_type(2))) float v2f;
typedef __attribute__((ext_vector_type(8))) float v8f;

#define H64 64

// Order-preserving float->uint key for atomicMax-based segment max.
__device__ __forceinline__ unsigned fkey(float f) {
    unsigned u = __float_as_uint(f);
    return (u & 0x80000000u) ? ~u : (u | 0x80000000u);
}
__device__ __forceinline__ float fdec(unsigned u) {
    return (u & 0x80000000u) ? __uint_as_float(u & 0x7fffffffu)
                             : __uint_as_float(~u);
}

// ---------------------------------------------------------------------------
// 4-in-1 GEMM: O{0..3} = X @ W{0..3} + b{0..3}, X: (N,64), W: (64,64).
// One wave (32 threads) per 16x16 output tile, K=64 in steps of 4 via
// V_WMMA_F32_16X16X4_F32.
//   A 16x4 layout : lanes0-15 {v0:K=0, v1:K=1}, lanes16-31 {v0:K=2, v1:K=3}
//   B 4x16 layout : mirrored striping (row K across lanes within a VGPR)
//   C/D 16x16     : vgpr i = row (i) lanes0-15, row (i+8) lanes16-31
// ---------------------------------------------------------------------------
__global__ void __launch_bounds__(32)
gemm4_wmma(const float* __restrict__ X,
           const float* __restrict__ W0, const float* __restrict__ B0, float* __restrict__ O0,
           const float* __restrict__ W1, const float* __restrict__ B1, float* __restrict__ O1,
           const float* __restrict__ W2, const float* __restrict__ B2, float* __restrict__ O2,
           const float* __restrict__ W3, const float* __restrict__ B3, float* __restrict__ O3)
{
    const int tile = blockIdx.x;
    const int rb   = tile >> 4;   // 16-row block of X
    const int sub  = tile & 15;
    const int mat  = sub >> 2;    // which of the 4 GEMMs
    const int cb   = sub & 3;     // 16-col block of the 64-wide output

    const float* W; const float* B; float* O;
    switch (mat) {
        case 0:  W = W0; B = B0; O = O0; break;
        case 1:  W = W1; B = B1; O = O1; break;
        case 2:  W = W2; B = B2; O = O2; break;
        default: W = W3; B = B3; O = O3; break;
    }

    const int lane = threadIdx.x & 31;
    const int half = lane >> 4;   // 0: lanes 0-15, 1: lanes 16-31
    const int l    = lane & 15;

    const float* Xrow = X + (size_t)(rb * 16 + l) * H64;
    const float* Wcol = W + cb * 16 + l;

    v8f acc = {};
    #pragma unroll
    for (int k0 = 0; k0 < H64; k0 += 4) {
        const int ka = k0 + half * 2;
        v2f a, b;
        a.x = Xrow[ka];
        a.y = Xrow[ka + 1];
        b.x = Wcol[(size_t)ka * H64];
        b.y = Wcol[(size_t)(ka + 1) * H64];
        acc = __builtin_amdgcn_wmma_f32_16x16x4_f32(
                  false, a, false, b, (short)0, acc, false, false);
    }

    const float bias = B[cb * 16 + l];
    #pragma unroll
    for (int i = 0; i < 8; ++i)
        O[(size_t)(rb * 16 + half * 8 + i) * H64 + cb * 16 + l] = acc[i] + bias;
}

// Scalar tail for N not divisible by 16 (not used for N=50000, kept for safety).
__global__ void gemm_tail(const float* __restrict__ X, const float* __restrict__ W,
                          const float* __restrict__ B, float* __restrict__ O,
                          int row0, int nrows)
{
    int idx = blockIdx.x * blockDim.x + threadIdx.x;
    if (idx >= nrows * H64) return;
    int row = row0 + (idx >> 6);
    int col = idx & 63;
    float acc = B[col];
    for (int k = 0; k < H64; ++k)
        acc += X[(size_t)row * H64 + k] * W[(size_t)k * H64 + col];
    O[(size_t)row * H64 + col] = acc;
}

// ---------------------------------------------------------------------------
// Edge phase. One wave (32 lanes) per edge; each lane covers features
// {lane, lane+32} -> fully coalesced 256B row reads.
// ---------------------------------------------------------------------------
__global__ void edge_logits(const int* __restrict__ src, const int* __restrict__ dst,
                            const float* __restrict__ ew,
                            const float* __restrict__ q, const float* __restrict__ k,
                            const float* __restrict__ We,
                            float* __restrict__ la, unsigned* __restrict__ mu, int Ecnt)
{
    int wave = blockIdx.x * (blockDim.x >> 5) + (threadIdx.x >> 5);
    int lane = threadIdx.x & 31;
    if (wave >= Ecnt) return;                 // uniform per wave
    int sN = src[wave], dN = dst[wave];
    float w = ew[wave];
    const float* qd = q + (size_t)dN * H64;
    const float* ks = k + (size_t)sN * H64;
    float p = qd[lane]      * (ks[lane]      + w * We[lane])
            + qd[lane + 32] * (ks[lane + 32] + w * We[lane + 32]);
    #pragma unroll
    for (int off = 16; off; off >>= 1) p += __shfl_xor(p, off, 32);
    if (lane == 0) {
        float logit = p * 0.125f;             // 1/sqrt(H), H=64
        la[wave] = logit;
        atomicMax(mu + dN, fkey(logit));
    }
}

__global__ void edge_exp(const int* __restrict__ dst, const float* __restrict__ la,
                         const unsigned* __restrict__ mu,
                         float* __restrict__ a, float* __restrict__ s, int Ecnt)
{
    int e = blockIdx.x * blockDim.x + threadIdx.x;
    if (e >= Ecnt) return;
    int dN = dst[e];
    float m  = fdec(mu[dN]);
    float av = expf(la[e] - m);
    a[e] = av;
    atomicAdd(s + dN, av);
}

__global__ void edge_agg(const int* __restrict__ src, const int* __restrict__ dst,
                         const float* __restrict__ ew,
                         const float* __restrict__ v, const float* __restrict__ We,
                         const float* __restrict__ a, const float* __restrict__ s,
                         float* __restrict__ agg, int Ecnt)
{
    int wave = blockIdx.x * (blockDim.x >> 5) + (threadIdx.x >> 5);
    int lane = threadIdx.x & 31;
    if (wave >= Ecnt) return;
    int sN = src[wave], dN = dst[wave];
    float w    = ew[wave];
    float coef = a[wave] / (s[dN] + 1e-16f);
    const float* vs = v + (size_t)sN * H64;
    float* ad = agg + (size_t)dN * H64;
    atomicAdd(ad + lane,      coef * (vs[lane]      + w * We[lane]));
    atomicAdd(ad + lane + 32, coef * (vs[lane + 32] + w * We[lane + 32]));
}

__global__ void node_combine(const float* __restrict__ agg, const float* __restrict__ sl,
                             float* __restrict__ hout, int n64)
{
    int idx = blockIdx.x * blockDim.x + threadIdx.x;
    if (idx >= n64) return;
    hout[idx] = fmaxf(agg[idx] + sl[idx], 0.0f);
}

// ---------------------------------------------------------------------------
// Pooling + classifier
// ---------------------------------------------------------------------------
__global__ void pool_kernel(const float* __restrict__ hbuf, const int* __restrict__ bat,
                            float* __restrict__ gsum, float* __restrict__ gcnt, int n64)
{
    int idx = blockIdx.x * blockDim.x + threadIdx.x;
    if (idx >= n64) return;
    int n = idx >> 6, h = idx & 63;
    int g = bat[n];
    atomicAdd(gsum + (size_t)g * H64 + h, hbuf[idx]);
    if (h == 0) atomicAdd(gcnt + g, 1.0f);
}

__global__ void classify(const float* __restrict__ gsum, const float* __restrict__ gcnt,
                         const float* __restrict__ Wl, const float* __restrict__ bl,
                         float* __restrict__ out, int G, int C)
{
    int idx = blockIdx.x * blockDim.x + threadIdx.x;
    if (idx >= G * C) return;
    int g = idx / C, c = idx % C;
    float cnt = fmaxf(gcnt[g], 1.0f);
    const float* gs = gsum + (size_t)g * H64;
    float dot = 0.0f;
    #pragma unroll
    for (int h = 0; h < H64; ++h) dot += gs[h] * Wl[h * C + c];
    out[idx] = bl[c] + dot / cnt;
}

// ---------------------------------------------------------------------------
// Host launcher
// ---------------------------------------------------------------------------
extern "C" void kernel_launch(void* const* d_in, const int* in_sizes, int n_in,
                              void* d_out, int out_size, void* d_ws, size_t ws_size,
                              hipStream_t stream)
{
    const float* x   = (const float*)d_in[0];
    const int*   ei  = (const int*)  d_in[1];
    const float* ew  = (const float*)d_in[2];
    const int*   bat = (const int*)  d_in[3];
    const float* Wq1 = (const float*)d_in[4];  const float* bq1 = (const float*)d_in[5];
    const float* Wk1 = (const float*)d_in[6];  const float* bk1 = (const float*)d_in[7];
    const float* Wv1 = (const float*)d_in[8];  const float* bv1 = (const float*)d_in[9];
    const float* We1 = (const float*)d_in[10];
    const float* Ws1 = (const float*)d_in[11]; const float* bs1 = (const float*)d_in[12];
    const float* Wq2 = (const float*)d_in[13]; const float* bq2 = (const float*)d_in[14];
    const float* Wk2 = (const float*)d_in[15]; const float* bk2 = (const float*)d_in[16];
    const float* Wv2 = (const float*)d_in[17]; const float* bv2 = (const float*)d_in[18];
    const float* We2 = (const float*)d_in[19];
    const float* Ws2 = (const float*)d_in[20]; const float* bs2 = (const float*)d_in[21];
    const float* Wl  = (const float*)d_in[22]; const float* bl  = (const float*)d_in[23];

    const int N = in_sizes[0] / H64;      // 50000
    const int E = in_sizes[2];            // 800000 (edge_weight length)
    const int C = in_sizes[22] / H64;     // 2
    const int G = out_size / C;           // 256
    const int* src = ei;
    const int* dst = ei + E;

    const size_t N64 = (size_t)N * H64;
    float* ws  = (float*)d_ws;
    float* q   = ws;
    float* k   = q  + N64;
    float* v   = k  + N64;
    float* sl  = v  + N64;
    float* agg = sl + N64;
    float* h1  = agg + N64;
    float* la  = h1 + N64;
    float* a   = la + (size_t)E;
    float* s   = a  + (size_t)E;
    unsigned* mu = (unsigned*)(s + N);
    float* gsum  = (float*)(mu + N);
    float* gcnt  = gsum + (size_t)G * H64;

    const int n16      = N / 16;
    const int nrem     = N - n16 * 16;
    const int ewaveBlk = (E + 7) / 8;          // 8 waves of 32 per 256-thread block
    const int ethrBlk  = (E + 255) / 256;
    const int nodeBlk  = (int)((N64 + 255) / 256);

    auto layer = [&](const float* X,
                     const float* Wq, const float* bq, const float* Wk, const float* bk,
                     const float* Wv, const float* bv, const float* We,
                     const float* Wsk, const float* bsk, float* hout)
    {
        hipMemsetAsync(agg, 0, N64 * sizeof(float), stream);
        hipMemsetAsync(s,   0, (size_t)N * sizeof(float), stream);
        hipMemsetAsync(mu,  0, (size_t)N * sizeof(unsigned), stream);

        if (n16 > 0)
            gemm4_wmma<<<n16 * 16, 32, 0, stream>>>(X, Wq, bq, q, Wk, bk, k,
                                                    Wv, bv, v, Wsk, bsk, sl);
        if (nrem > 0) {
            int tb = (nrem * H64 + 255) / 256;
            gemm_tail<<<tb, 256, 0, stream>>>(X, Wq, bq,  q,  n16 * 16, nrem);
            gemm_tail<<<tb, 256, 0, stream>>>(X, Wk, bk,  k,  n16 * 16, nrem);
            gemm_tail<<<tb, 256, 0, stream>>>(X, Wv, bv,  v,  n16 * 16, nrem);
            gemm_tail<<<tb, 256, 0, stream>>>(X, Wsk, bsk, sl, n16 * 16, nrem);
        }

        edge_logits<<<ewaveBlk, 256, 0, stream>>>(src, dst, ew, q, k, We, la, mu, E);
        edge_exp  <<<ethrBlk,  256, 0, stream>>>(dst, la, mu, a, s, E);
        edge_agg  <<<ewaveBlk, 256, 0, stream>>>(src, dst, ew, v, We, a, s, agg, E);
        node_combine<<<nodeBlk, 256, 0, stream>>>(agg, sl, hout, (int)N64);
    };

    layer(x,  Wq1, bq1, Wk1, bk1, Wv1, bv1, We1, Ws1, bs1, h1);
    layer(h1, Wq2, bq2, Wk2, bk2, Wv2, bv2, We2, Ws2, bs2, h1);

    hipMemsetAsync(gsum, 0, (size_t)G * H64 * sizeof(float), stream);
    hipMemsetAsync(gcnt, 0, (size_t)G * sizeof(float), stream);
    pool_kernel<<<nodeBlk, 256, 0, stream>>>(h1, bat, gsum, gcnt, (int)N64);
    classify<<<(G * C + 255) / 256, 256, 0, stream>>>(gsum, gcnt, Wl, bl,
                                                      (float*)d_out, G, C);
}